// DevignModel_2542620639584
// MI455X (gfx1250) — compile-verified
//
#include <hip/hip_runtime.h>
#include <hip/hip_bf16.h>

typedef __attribute__((ext_vector_type(2))) float v2f;
typedef __attribute__((ext_vector_type(8))) float v8f;

#define DN 128
#define CN 256
#define NNODES 65536
#define NG 128
#define LL 512
#define NE 262144

// ---------------------------------------------------------------------------
// Generic WMMA f32 GEMM: C[M,Ncols] = A[M,K] * B (+ per-column bias)
// transB==0: B is [K,Ncols] row-major.  transB==1: B is [Ncols,K] row-major.
// Block = 256 threads = 8 waves, block tile 128x128, wave tile 32x64.
// All dims are multiples of the tile sizes for every call site.
// ---------------------------------------------------------------------------
__global__ __launch_bounds__(256) void wmma_gemm_kernel(
    const float* __restrict__ A, const float* __restrict__ B,
    const float* __restrict__ bias, float* __restrict__ C,
    int M, int Ncols, int K, int transB)
{
  const int lane = threadIdx.x & 31;
  const int wave = threadIdx.x >> 5;
  const int wm = wave >> 1, wn = wave & 1;
  const int m0 = blockIdx.y * 128 + wm * 32;
  const int n0 = blockIdx.x * 128 + wn * 64;
  const int lm = lane & 15;
  const int khalf = (lane >> 4) << 1;   // A/B K-pair held by this lane half
  v8f acc[2][4] = {};
  for (int k0 = 0; k0 < K; k0 += 4) {
    const int kk = k0 + khalf;
    v2f a[2], b[4];
#pragma unroll
    for (int mt = 0; mt < 2; ++mt) {
      const float* ap = A + (size_t)(m0 + mt * 16 + lm) * K + kk;
      a[mt].x = ap[0]; a[mt].y = ap[1];
    }
#pragma unroll
    for (int nt = 0; nt < 4; ++nt) {
      const int col = n0 + nt * 16 + lm;
      if (transB) {
        const float* bp = B + (size_t)col * K + kk;
        b[nt].x = bp[0]; b[nt].y = bp[1];
      } else {
        const float* bp = B + (size_t)kk * Ncols + col;
        b[nt].x = bp[0]; b[nt].y = bp[Ncols];
      }
    }
#pragma unroll
    for (int mt = 0; mt < 2; ++mt)
#pragma unroll
      for (int nt = 0; nt < 4; ++nt)
        acc[mt][nt] = __builtin_amdgcn_wmma_f32_16x16x4_f32(
            false, a[mt], false, b[nt], (short)0, acc[mt][nt], false, false);
  }
  const int rowhalf = (lane >> 4) << 3;
#pragma unroll
  for (int mt = 0; mt < 2; ++mt) {
#pragma unroll
    for (int nt = 0; nt < 4; ++nt) {
      const int col = n0 + nt * 16 + lm;
      const float bv = bias ? bias[col] : 0.0f;
#pragma unroll
      for (int v = 0; v < 8; ++v) {
        const int row = m0 + mt * 16 + rowhalf + v;
        C[(size_t)row * Ncols + col] = acc[mt][nt][v] + bv;
      }
    }
  }
}

// ---------------------------------------------------------------------------
// Implicit-GEMM Conv1d (VALID) with WMMA f32.
// W: repacked weights [Cout, Ktot] with kk = kpos*Cin + ci (contiguous ci).
// in: channel-major [g_stride per graph][Cin rows of in_ld floats].
// out[g, row, col] = bias[row] + sum_kk W[row,kk] * in[g, ci, col+kpos]
// Stores masked to col < Lout (out row stride = out_ld).
// ---------------------------------------------------------------------------
__global__ __launch_bounds__(256) void conv_wmma_kernel(
    const float* __restrict__ W, const float* __restrict__ bias,
    const float* __restrict__ in, float* __restrict__ out,
    int Cin, int Cout, int Ktot, int in_ld, size_t g_stride,
    int Lout, int out_ld)
{
  const int g = blockIdx.z;
  const int lane = threadIdx.x & 31;
  const int wave = threadIdx.x >> 5;
  const int wm = wave >> 1, wn = wave & 1;
  const int m0 = blockIdx.y * 128 + wm * 32;
  const int n0 = blockIdx.x * 128 + wn * 64;
  const int lm = lane & 15;
  const int khalf = (lane >> 4) << 1;
  const float* gin = in + (size_t)g * g_stride;
  v8f acc[2][4] = {};
  for (int k0 = 0; k0 < Ktot; k0 += 4) {
    const int kk = k0 + khalf;          // kk and kk+1 share kpos (Cin % 4 == 0)
    const int kpos = kk / Cin;
    const int ci = kk - kpos * Cin;
    v2f a[2], b[4];
#pragma unroll
    for (int mt = 0; mt < 2; ++mt) {
      const float* ap = W + (size_t)(m0 + mt * 16 + lm) * Ktot + kk;
      a[mt].x = ap[0]; a[mt].y = ap[1];
    }
#pragma unroll
    for (int nt = 0; nt < 4; ++nt) {
      const int col = n0 + nt * 16 + lm;
      const float* bp = gin + (size_t)ci * in_ld + col + kpos;
      b[nt].x = bp[0]; b[nt].y = bp[in_ld];
    }
#pragma unroll
    for (int mt = 0; mt < 2; ++mt)
#pragma unroll
      for (int nt = 0; nt < 4; ++nt)
        acc[mt][nt] = __builtin_amdgcn_wmma_f32_16x16x4_f32(
            false, a[mt], false, b[nt], (short)0, acc[mt][nt], false, false);
  }
  const int rowhalf = (lane >> 4) << 3;
#pragma unroll
  for (int mt = 0; mt < 2; ++mt) {
#pragma unroll
    for (int nt = 0; nt < 4; ++nt) {
      const int col = n0 + nt * 16 + lm;
      if (col < Lout) {
#pragma unroll
        for (int v = 0; v < 8; ++v) {
          const int row = m0 + mt * 16 + rowhalf + v;
          out[((size_t)g * Cout + row) * out_ld + col] = acc[mt][nt][v] + bias[row];
        }
      }
    }
  }
}

// Weighted message scatter-add: one wave per edge, 4 channels per lane.
__global__ __launch_bounds__(256) void edge_scatter_kernel(
    const float* __restrict__ m, const int* __restrict__ src,
    const int* __restrict__ dst, const float* __restrict__ ew,
    float* __restrict__ agg)
{
  const int gid = blockIdx.x * blockDim.x + threadIdx.x;
  const int e = gid >> 5;
  if (e >= NE) return;
  const int q = (gid & 31) << 2;
  const int s = src[e];
  const int d = dst[e];
  const float w = ew[e];
  const float4 mv = *(const float4*)(m + (size_t)s * DN + q);
  float* ap = agg + (size_t)d * DN + q;
  atomicAdd(ap + 0, mv.x * w);
  atomicAdd(ap + 1, mv.y * w);
  atomicAdd(ap + 2, mv.z * w);
  atomicAdd(ap + 3, mv.w * w);
}

// GRU cell elementwise update (gates precomputed by WMMA GEMMs).
__global__ __launch_bounds__(256) void gru_kernel(
    const float* __restrict__ gx, const float* __restrict__ gh,
    float* __restrict__ h)
{
  const int idx = blockIdx.x * blockDim.x + threadIdx.x;
  if (idx >= NNODES * DN) return;
  const int n = idx >> 7;
  const int d = idx & 127;
  const float* gxr = gx + (size_t)n * (3 * DN);
  const float* ghr = gh + (size_t)n * (3 * DN);
  const float r = 1.0f / (1.0f + __expf(-(gxr[d] + ghr[d])));
  const float z = 1.0f / (1.0f + __expf(-(gxr[DN + d] + ghr[DN + d])));
  const float nn = tanhf(gxr[2 * DN + d] + r * ghr[2 * DN + d]);
  h[idx] = (1.0f - z) * nn + z * h[idx];
}

// Build channel-major concat tensor cT[g, c, l]; c<128 -> h, else x. LDS-tiled.
__global__ __launch_bounds__(256) void concat_transpose_kernel(
    const float* __restrict__ h, const float* __restrict__ x,
    float* __restrict__ cT)
{
  __shared__ float tile[32][33];
  const int g = blockIdx.z;
  const int l0 = blockIdx.x * 32;
  const int c0 = blockIdx.y * 32;
  const int c = c0 + threadIdx.x;
  for (int r = threadIdx.y; r < 32; r += 8) {
    const int node = g * LL + l0 + r;
    float v;
    if (c < DN) v = h[(size_t)node * DN + c];
    else        v = x[(size_t)node * DN + (c - DN)];
    tile[r][threadIdx.x] = v;
  }
  __syncthreads();
  for (int r = threadIdx.y; r < 32; r += 8) {
    const int cc = c0 + r;
    const int l = l0 + threadIdx.x;
    cT[((size_t)g * CN + cc) * LL + l] = tile[threadIdx.x][r];
  }
}

// Repack conv weights [Cout,Cin,ks] -> [Cout, ks*Cin] with kk = k*Cin + i.
__global__ void repack_w_kernel(const float* __restrict__ w,
                                float* __restrict__ wr,
                                int Cout, int Cin, int ks)
{
  const int idx = blockIdx.x * blockDim.x + threadIdx.x;
  const int K = Cin * ks;
  if (idx >= Cout * K) return;
  const int o = idx / K;
  const int rem = idx - o * K;
  const int k = rem / Cin;
  const int i = rem - k * Cin;
  wr[idx] = w[((size_t)o * Cin + i) * ks + k];
}

// Training-mode BatchNorm statistics per channel over (G, Lvalid).
__global__ __launch_bounds__(256) void bn_stats_kernel(
    const float* __restrict__ in, float* __restrict__ mean,
    float* __restrict__ istd, int C, int ld, int Lvalid)
{
  const int c = blockIdx.x;
  __shared__ float ss[256], ss2[256];
  float s = 0.f, s2 = 0.f;
  const int total = NG * Lvalid;
  for (int idx = threadIdx.x; idx < total; idx += 256) {
    const int g = idx / Lvalid;
    const int l = idx - g * Lvalid;
    const float v = in[((size_t)g * C + c) * ld + l];
    s += v; s2 += v * v;
  }
  ss[threadIdx.x] = s; ss2[threadIdx.x] = s2;
  __syncthreads();
  for (int o = 128; o > 0; o >>= 1) {
    if (threadIdx.x < o) {
      ss[threadIdx.x] += ss[threadIdx.x + o];
      ss2[threadIdx.x] += ss2[threadIdx.x + o];
    }
    __syncthreads();
  }
  if (threadIdx.x == 0) {
    const float inv = 1.0f / (float)total;
    const float m = ss[0] * inv;
    const float var = fmaxf(ss2[0] * inv - m * m, 0.f);
    mean[c] = m;
    istd[c] = rsqrtf(var + 1e-5f);
  }
}

// Fused BN-apply + ReLU + max-pool (kernel pk, stride ps).
__global__ __launch_bounds__(256) void bn_relu_pool_kernel(
    const float* __restrict__ in, const float* __restrict__ mean,
    const float* __restrict__ istd, const float* __restrict__ gamma,
    const float* __restrict__ beta, float* __restrict__ out,
    int C, int in_ld, int pk, int ps, int Lpool, int out_ld)
{
  const int idx = blockIdx.x * blockDim.x + threadIdx.x;
  const int total = NG * C * Lpool;
  if (idx >= total) return;
  const int p = idx % Lpool;
  const int c = (idx / Lpool) % C;
  const int g = idx / (Lpool * C);
  const float sc = istd[c] * gamma[c];
  const float sh = beta[c] - mean[c] * sc;
  const float* ip = in + ((size_t)g * C + c) * in_ld + p * ps;
  float mx = 0.0f;  // relu makes every candidate >= 0
  for (int j = 0; j < pk; ++j)
    mx = fmaxf(mx, fmaxf(ip[j] * sc + sh, 0.0f));
  out[((size_t)g * C + c) * out_ld + p] = mx;
}

// out[g,j] = mean_p ( (Y2[g,:,p] . wy[j] + by[j]) * (Z2[g,:,p] . wz[j] + bz[j]) )
__global__ __launch_bounds__(128) void final_kernel(
    const float* __restrict__ Y2, const float* __restrict__ Z2,
    const float* __restrict__ wy, const float* __restrict__ by,
    const float* __restrict__ wz, const float* __restrict__ bz,
    float* __restrict__ out)
{
  const int g = blockIdx.x;
  const int p = threadIdx.x;
  __shared__ float s0[128], s1[128];
  float p0 = 0.f, p1 = 0.f;
  if (p < 127) {
    float y0 = by[0], y1 = by[1], z0 = bz[0], z1 = bz[1];
    for (int d = 0; d < DN; ++d) {
      const float v = Y2[((size_t)g * DN + d) * 128 + p];
      y0 += v * wy[d];
      y1 += v * wy[DN + d];
    }
    for (int c = 0; c < CN; ++c) {
      const float v = Z2[((size_t)g * CN + c) * 128 + p];
      z0 += v * wz[c];
      z1 += v * wz[CN + c];
    }
    p0 = y0 * z0; p1 = y1 * z1;
  }
  s0[p] = p0; s1[p] = p1;
  __syncthreads();
  for (int o = 64; o > 0; o >>= 1) {
    if (p < o) { s0[p] += s0[p + o]; s1[p] += s1[p + o]; }
    __syncthreads();
  }
  if (p == 0) {
    out[g * 2 + 0] = s0[0] / 127.0f;
    out[g * 2 + 1] = s1[0] / 127.0f;
  }
}

extern "C" void kernel_launch(void* const* d_in, const int* in_sizes, int n_in,
                              void* d_out, int out_size, void* d_ws, size_t ws_size,
                              hipStream_t stream)
{
  (void)in_sizes; (void)n_in; (void)out_size; (void)ws_size;
  const float* x        = (const float*)d_in[0];
  const int*   ei       = (const int*)d_in[1];
  const float* ew       = (const float*)d_in[2];
  // d_in[3] = batch: arange(N)//L, structure is implicit (pure reshape)
  const float* ggnn_w   = (const float*)d_in[4];
  const float* gru_wih  = (const float*)d_in[5];
  const float* gru_whh  = (const float*)d_in[6];
  const float* gru_bih  = (const float*)d_in[7];
  const float* gru_bhh  = (const float*)d_in[8];
  const float* conv1_w  = (const float*)d_in[9];
  const float* conv1_b  = (const float*)d_in[10];
  const float* conv2_w  = (const float*)d_in[11];
  const float* conv2_b  = (const float*)d_in[12];
  const float* convc1_w = (const float*)d_in[13];
  const float* convc1_b = (const float*)d_in[14];
  const float* convc2_w = (const float*)d_in[15];
  const float* convc2_b = (const float*)d_in[16];
  const float* bn1_g    = (const float*)d_in[17];
  const float* bn1_b    = (const float*)d_in[18];
  const float* bn2_g    = (const float*)d_in[19];
  const float* bn2_b    = (const float*)d_in[20];
  const float* mlpy_w   = (const float*)d_in[21];
  const float* mlpy_b   = (const float*)d_in[22];
  const float* mlpz_w   = (const float*)d_in[23];
  const float* mlpz_b   = (const float*)d_in[24];
  float* out = (float*)d_out;

  const size_t MB = 1ull << 20;
  char* ws = (char*)d_ws;
  float* hbuf   = (float*)(ws + 0 * MB);    // 32 MB, persistent
  float* mbuf   = (float*)(ws + 32 * MB);   // 32 MB
  float* aggbuf = (float*)(ws + 64 * MB);   // 32 MB
  float* gxbuf  = (float*)(ws + 96 * MB);   // 96 MB
  float* ghbuf  = (float*)(ws + 192 * MB);  // 96 MB
  // post-GGNN reuse (hbuf stays live):
  float* cT     = (float*)(ws + 32 * MB);   // 67 MB [G,256,512]
  float* Yc1    = (float*)(ws + 100 * MB);  // 32 MB [G,128,512]
  float* Ypool1 = (float*)(ws + 132 * MB);  // 16 MB [G,128,256]
  float* Yc2    = (float*)(ws + 148 * MB);  // 16 MB [G,128,256]
  float* Ypool2 = (float*)(ws + 164 * MB);  //  8 MB [G,128,128]
  float* Zc1    = (float*)(ws + 172 * MB);  // 64 MB [G,256,512]
  float* Zpool1 = (float*)(ws + 236 * MB);  // 32 MB [G,256,256]
  float* Zc2    = (float*)(ws + 100 * MB);  // 32 MB (Yc1 dead)
  float* Zpool2 = (float*)(ws + 132 * MB);  // 16 MB (Ypool1 dead)
  float* wr1    = (float*)(ws + 288 * MB);  // 192 KB
  float* wrc1   = (float*)(ws + 289 * MB);  // 768 KB
  float* meanb  = (float*)(ws + 290 * MB);
  float* istdb  = (float*)(ws + 290 * MB + 4096);

  // h = x
  hipMemcpyAsync(hbuf, x, (size_t)NNODES * DN * sizeof(float),
                 hipMemcpyDeviceToDevice, stream);

  // ---- GatedGraphConv: 6 steps of linear -> scatter-add -> GRU -------------
  for (int s = 0; s < 6; ++s) {
    // m = h @ W_s   (NN)
    wmma_gemm_kernel<<<dim3(1, 512), 256, 0, stream>>>(
        hbuf, ggnn_w + (size_t)s * DN * DN, nullptr, mbuf, NNODES, DN, DN, 0);
    hipMemsetAsync(aggbuf, 0, (size_t)NNODES * DN * sizeof(float), stream);
    edge_scatter_kernel<<<(NE * 32) / 256, 256, 0, stream>>>(
        mbuf, ei, ei + NE, ew, aggbuf);
    // gx = agg @ wih^T + bih ; gh = h @ whh^T + bhh   (NT)
    wmma_gemm_kernel<<<dim3(3, 512), 256, 0, stream>>>(
        aggbuf, gru_wih, gru_bih, gxbuf, NNODES, 3 * DN, DN, 1);
    wmma_gemm_kernel<<<dim3(3, 512), 256, 0, stream>>>(
        hbuf, gru_whh, gru_bhh, ghbuf, NNODES, 3 * DN, DN, 1);
    gru_kernel<<<(NNODES * DN) / 256, 256, 0, stream>>>(gxbuf, ghbuf, hbuf);
  }

  // ---- de-batchify + concat, channel-major ---------------------------------
  concat_transpose_kernel<<<dim3(16, 8, NG), dim3(32, 8), 0, stream>>>(hbuf, x, cT);
  repack_w_kernel<<<(128 * 384) / 256, 256, 0, stream>>>(conv1_w, wr1, 128, 128, 3);
  repack_w_kernel<<<(256 * 768) / 256, 256, 0, stream>>>(convc1_w, wrc1, 256, 256, 3);

  // ---- Y branch (D=128 channels, h only) -----------------------------------
  conv_wmma_kernel<<<dim3(4, 1, NG), 256, 0, stream>>>(
      wr1, conv1_b, cT, Yc1, 128, 128, 384, 512, (size_t)256 * 512, 510, 512);
  bn_stats_kernel<<<128, 256, 0, stream>>>(Yc1, meanb, istdb, 128, 512, 510);
  bn_relu_pool_kernel<<<(NG * 128 * 254 + 255) / 256, 256, 0, stream>>>(
      Yc1, meanb, istdb, bn1_g, bn1_b, Ypool1, 128, 512, 3, 2, 254, 256);
  conv_wmma_kernel<<<dim3(2, 1, NG), 256, 0, stream>>>(
      conv2_w, conv2_b, Ypool1, Yc2, 128, 128, 128, 256, (size_t)128 * 256, 254, 256);
  bn_stats_kernel<<<128, 256, 0, stream>>>(Yc2, meanb, istdb, 128, 256, 254);
  bn_relu_pool_kernel<<<(NG * 128 * 127 + 255) / 256, 256, 0, stream>>>(
      Yc2, meanb, istdb, bn1_g, bn1_b, Ypool2, 128, 256, 2, 2, 127, 128);

  // ---- Z branch (C=256 channels, concat[h,x]) ------------------------------
  conv_wmma_kernel<<<dim3(4, 2, NG), 256, 0, stream>>>(
      wrc1, convc1_b, cT, Zc1, 256, 256, 768, 512, (size_t)256 * 512, 510, 512);
  bn_stats_kernel<<<256, 256, 0, stream>>>(Zc1, meanb, istdb, 256, 512, 510);
  bn_relu_pool_kernel<<<(NG * 256 * 254 + 255) / 256, 256, 0, stream>>>(
      Zc1, meanb, istdb, bn2_g, bn2_b, Zpool1, 256, 512, 3, 2, 254, 256);
  conv_wmma_kernel<<<dim3(2, 2, NG), 256, 0, stream>>>(
      convc2_w, convc2_b, Zpool1, Zc2, 256, 256, 256, 256, (size_t)256 * 256, 254, 256);
  bn_stats_kernel<<<256, 256, 0, stream>>>(Zc2, meanb, istdb, 256, 256, 254);
  bn_relu_pool_kernel<<<(NG * 256 * 127 + 255) / 256, 256, 0, stream>>>(
      Zc2, meanb, istdb, bn2_g, bn2_b, Zpool2, 256, 256, 2, 2, 127, 128);

  // ---- readout -------------------------------------------------------------
  final_kernel<<<NG, 128, 0, stream>>>(Ypool2, Zpool2, mlpy_w, mlpy_b,
                                       mlpz_w, mlpz_b, out);
}